// ContrastiveMaskedPatchSimilarity_69913477644540
// MI455X (gfx1250) — compile-verified
//
#include <hip/hip_runtime.h>

// Problem constants (from the reference).
#define BB   32
#define CC   256
#define HH   64
#define WW   64
#define HW   (HH * WW)        // 4096
#define NPOS (BB * HW)        // 131072 spatial positions
#define EPSV 1e-8f

typedef float v2f __attribute__((ext_vector_type(2)));
typedef float v8f __attribute__((ext_vector_type(8)));

// ---------------------------------------------------------------------------
// Main kernel: each wave32 owns 16 consecutive spatial positions of one batch
// image (HW=4096 is a multiple of 16, so a wave never crosses a batch edge).
//
// V_WMMA_F32_16X16X4_F32 diagonal trick:
//   A (16x4)  : A[p][k] = u[b, cbase+k, posbase+p]
//   B (4x16)  : B[k][p] = m[b, cbase+k, posbase+p]
//   D[p][p]  += sum_k u[p,k] * m[p,k]        (accumulated over 64 K-chunks)
// Per the CDNA5 VGPR layout tables, A and B use the IDENTICAL per-lane
// mapping for this shape: lane L holds (pos = L%16, K = 2*(L/16) + {0,1})
// in its two operand VGPRs. So lanes 0-15 / 16-31 each load a contiguous,
// fully-consumed 64B segment, and the same registers feed A and B:
//   num += U x M,  uu += U x U,  mm += M x M   -> exact f32 dot products.
// ---------------------------------------------------------------------------
__global__ __launch_bounds__(256) void cms_main(const float* __restrict__ u,
                                                const float* __restrict__ m,
                                                const int*   __restrict__ msk,
                                                float*       __restrict__ acc) {
  const int lane   = threadIdx.x & 31;
  const int waveId = blockIdx.x * (blockDim.x >> 5) + (threadIdx.x >> 5);
  const int posBase = waveId * 16;            // 16 positions per wave
  const int b  = posBase / HW;
  const int sp = posBase % HW;                // spatial offset within image
  const int p16 = lane & 15;                  // this lane's position slot
  const int co  = (lane >> 4) << 1;           // channel offset 0 or 2 (K pair)

  const size_t base = ((size_t)b * CC + co) * HW + (size_t)sp + p16;
  const float* up = u + base;
  const float* mp = m + base;

  v8f numA = {0.f, 0.f, 0.f, 0.f, 0.f, 0.f, 0.f, 0.f};
  v8f uuA  = numA;
  v8f mmA  = numA;

  // 256 channels in chunks of 4 (K=4 per WMMA). Constant strides let the
  // compiler fold everything into IOFFSET immediates off one base address.
#pragma unroll 4
  for (int i = 0; i < CC / 4; ++i) {
    const size_t o0 = (size_t)i * 4 * HW;     // advance 4 channels
    v2f A, Bm;
    A.x  = up[o0];          // channel 4i+co   -> operand VGPR0 (K = co)
    A.y  = up[o0 + HW];     // channel 4i+co+1 -> operand VGPR1 (K = co+1)
    Bm.x = mp[o0];
    Bm.y = mp[o0 + HW];
    numA = __builtin_amdgcn_wmma_f32_16x16x4_f32(false, A,  false, Bm,
                                                 (short)0, numA, false, false);
    uuA  = __builtin_amdgcn_wmma_f32_16x16x4_f32(false, A,  false, A,
                                                 (short)0, uuA,  false, false);
    mmA  = __builtin_amdgcn_wmma_f32_16x16x4_f32(false, Bm, false, Bm,
                                                 (short)0, mmA,  false, false);
  }

  // Diagonal extraction. C/D layout: VGPR r holds row M=r on lanes 0-15
  // (col = lane) and row M=r+8 on lanes 16-31 (col = lane-16). So diag p is:
  //   p in 0..7  -> VGPR p,   lane p
  //   p in 8..15 -> VGPR p-8, lane p+16   (lanes 24..31)
  float nd = 0.f, ud = 0.f, md = 0.f;
  const bool isDiag = (lane < 8) || (lane >= 24);
#pragma unroll
  for (int r = 0; r < 8; ++r) {
    const bool pick = isDiag && ((lane & 7) == r);
    nd = pick ? numA[r] : nd;
    ud = pick ? uuA[r]  : ud;
    md = pick ? mmA[r]  : md;
  }

  float contrib = 0.f, cnt = 0.f;
  if (isDiag) {
    const int p  = (lane < 8) ? lane : (lane - 16);
    const int mk = msk[(size_t)b * HW + sp + p];
    if (mk != 0) {
      const float den = fmaxf(sqrtf(ud), EPSV) * fmaxf(sqrtf(md), EPSV);
      contrib = nd / den;
      cnt     = 1.f;
    }
  }

  // wave32 butterfly reduction (non-diagonal lanes contribute 0).
#pragma unroll
  for (int off = 16; off > 0; off >>= 1) {
    contrib += __shfl_xor(contrib, off, 32);
    cnt     += __shfl_xor(cnt,     off, 32);
  }
  if (lane == 0) {
    atomicAdd(&acc[0], contrib);
    atomicAdd(&acc[1], cnt);
  }
}

__global__ void cms_zero(float* __restrict__ acc) {
  if (threadIdx.x < 2) acc[threadIdx.x] = 0.f;
}

__global__ void cms_final(const float* __restrict__ acc,
                          float* __restrict__ out) {
  out[0] = acc[0] / acc[1];
}

extern "C" void kernel_launch(void* const* d_in, const int* in_sizes, int n_in,
                              void* d_out, int out_size, void* d_ws, size_t ws_size,
                              hipStream_t stream) {
  (void)in_sizes; (void)n_in; (void)out_size; (void)ws_size;
  const float* u   = (const float*)d_in[0];   // unmasked_latent_tensors
  const float* m   = (const float*)d_in[1];   // masked_latent_tensors
  const int*   msk = (const int*)d_in[2];     // latent_mask
  float* out = (float*)d_out;
  float* acc = (float*)d_ws;                  // [0]=sum(sim*mask), [1]=sum(mask)

  cms_zero<<<1, 32, 0, stream>>>(acc);

  const int waves  = NPOS / 16;               // 8192 waves
  const int blocks = waves / 8;               // 256 threads = 8 waves / block
  cms_main<<<blocks, 256, 0, stream>>>(u, m, msk, acc);

  cms_final<<<1, 1, 0, stream>>>(acc, out);
}